// SegmentedPolynomialNaive_49031346651201
// MI455X (gfx1250) — compile-verified
//
#include <hip/hip_runtime.h>

// ---------------------------------------------------------------------------
// Segmented tensor-product scatter:  out[dst] += x[src] (x) sh   (U=32, S=4)
// Memory-bound (≈141 MB HBM -> ~6 us floor @23.3 TB/s); out (51.2 MB) is kept
// L2-resident (192 MB L2) so the 410M-lane f32 atomic RMWs stay on-chip.
// CDNA5 path: double-buffered GLOBAL_LOAD_ASYNC_TO_LDS_B128 (ASYNCcnt) with
// non-temporal hint for the streamed sh/src/dst tiles, coalesced
// global_atomic_add_f32 (4 x contiguous 128B spans per edge).
// ---------------------------------------------------------------------------

#define AS1 __attribute__((address_space(1)))
#define AS3 __attribute__((address_space(3)))

typedef int v4i_t __attribute__((ext_vector_type(4)));

#if defined(__has_builtin)
#if __has_builtin(__builtin_amdgcn_global_load_async_to_lds_b128)
#define HAVE_ASYNC_B128 1
#endif
#if __has_builtin(__builtin_amdgcn_s_wait_asynccnt)
#define HAVE_WAIT_ASYNCCNT 1
#endif
#endif

// TH=NT (temporal hint: non-temporal) so streamed tiles don't evict the
// L2-resident output buffer.
#define CPOL_NT 1

__device__ __forceinline__ void async_copy16(const void* g, void* l) {
#if defined(HAVE_ASYNC_B128)
  __builtin_amdgcn_global_load_async_to_lds_b128((AS1 v4i_t*)g, (AS3 v4i_t*)l,
                                                 /*offset=*/0,
                                                 /*cpol=*/CPOL_NT);
#else
  asm volatile("global_load_async_to_lds_b128 %0, %1, off th:TH_LOAD_NT"
               :
               : "v"((AS3 v4i_t*)l), "v"((AS1 v4i_t*)g)
               : "memory");
#endif
}

template <int N>
__device__ __forceinline__ void wait_asynccnt() {
#if defined(HAVE_WAIT_ASYNCCNT)
  __builtin_amdgcn_s_wait_asynccnt(N);
#else
  asm volatile("s_wait_asynccnt %0" ::"i"(N) : "memory");
#endif
}

__device__ __forceinline__ void atomic_fadd(float* p, float v) {
  // Native global_atomic_add_f32 (no return -> STOREcnt, fire and forget).
  unsafeAtomicAdd(p, v);
}

static constexpr int TILE    = 1024;  // edges per LDS tile
static constexpr int THREADS = 256;   // 8 wave32
static constexpr int WAVES   = THREADS / 32;

__global__ __launch_bounds__(THREADS, 2) void tp_scatter_kernel(
    const float* __restrict__ x,    // [N,32]
    const float* __restrict__ sh,   // [E,4]
    const int* __restrict__ src,    // [E]
    const int* __restrict__ dst,    // [E]
    float* __restrict__ out,        // [N,128] (pre-zeroed)
    int E, int numTiles) {
  __shared__ float lsh[2][TILE * 4];       // 2 x 16 KB
  // src tile in [0,TILE), dst tile in [TILE,2*TILE): src[le] and dst[le]
  // are read by two separate LDS loads from the halves 4KB apart (no interleave).
  __shared__ int lidx[2][2 * TILE];        // 2 x  8 KB  (src | dst halves)

  const int tid = threadIdx.x;
  const int stride = gridDim.x;

  // Stage one tile of sh/src/dst into LDS buffer b. Returns true if it used
  // the async (ASYNCcnt) path: 6 outstanding wave-ops (4 sh + 1 src + 1 dst).
  auto issue = [&](int t, int b) -> bool {
    const long long base = (long long)t * TILE;
    long long rem = (long long)E - base;
    const int cnt = rem < TILE ? (int)rem : TILE;
    if (cnt == TILE) {
      const float* gsh = sh + base * 4;
#pragma unroll
      for (int j = 0; j < 4; ++j) {
        const int c = tid + j * THREADS;        // 16B chunk id (1024 total)
        async_copy16(gsh + c * 4, &lsh[b][c * 4]);
      }
      async_copy16(src + base + tid * 4, &lidx[b][tid * 4]);
      async_copy16(dst + base + tid * 4, &lidx[b][TILE + tid * 4]);
      return true;
    }
    // Tail tile: plain LDS fill (barrier before use covers visibility).
    for (int i = tid; i < cnt * 4; i += THREADS) lsh[b][i] = sh[base * 4 + i];
    for (int i = tid; i < cnt; i += THREADS) {
      lidx[b][i] = src[base + i];
      lidx[b][TILE + i] = dst[base + i];
    }
    return false;
  };

  int buf = 0;
  (void)issue(blockIdx.x, 0);

  for (int t = blockIdx.x; t < numTiles; t += stride, buf ^= 1) {
    // Prefetch next tile into the other buffer while current is consumed.
    const int nt = t + stride;
    bool nextAsync = false;
    if (nt < numTiles) nextAsync = issue(nt, buf ^ 1);

    // Wait for *current* tile's 6 async ops (async loads complete in order),
    // then make the LDS contents visible block-wide.
    if (nextAsync) wait_asynccnt<6>();
    else           wait_asynccnt<0>();
    __syncthreads();

    const long long base = (long long)t * TILE;
    long long rem = (long long)E - base;
    const int cnt = rem < TILE ? (int)rem : TILE;

    const int wave = tid >> 5;
    const int lane = tid & 31;

    // One wave per edge; interleaved so all 8 waves stay busy on tails.
#pragma unroll 2
    for (int le = wave; le < cnt; le += WAVES) {
      const int sidx = lidx[buf][le];
      const int didx = lidx[buf][TILE + le];

      // Prefetch a future gather row (x is L2-resident; cheap insurance).
      const int pe = le + 4 * WAVES;
      if (pe < cnt) __builtin_prefetch(&x[(long long)lidx[buf][pe] * 32], 0, 0);

      // Coalesced 128B row load: lane l holds x[sidx, l].
      const float xrow = x[(long long)sidx * 32 + lane];
      // sh broadcast within lane quads: lane l uses sh[le, l&3].
      const float shv = lsh[buf][le * 4 + (lane & 3)];

      float* o = out + (long long)didx * 128;
      // Output row element f = k*32+lane maps to (u = f>>2, s = f&3), so each
      // of the 4 atomic instructions covers one fully-contiguous 128B span.
#pragma unroll
      for (int k = 0; k < 4; ++k) {
        const float xv = __shfl(xrow, k * 8 + (lane >> 2), 32);
        atomic_fadd(o + k * 32 + lane, xv * shv);
      }
    }
    __syncthreads();  // buffer `buf` reused for tile t+2*stride
  }
}

extern "C" void kernel_launch(void* const* d_in, const int* in_sizes, int n_in,
                              void* d_out, int out_size, void* d_ws,
                              size_t ws_size, hipStream_t stream) {
  (void)n_in;
  (void)d_ws;
  (void)ws_size;
  const float* x = (const float*)d_in[0];
  const float* sh = (const float*)d_in[1];
  const int* src = (const int*)d_in[2];
  const int* dst = (const int*)d_in[3];
  float* out = (float*)d_out;

  const int E = in_sizes[2];

  // Zero-init the scatter target (harness poisons d_out).
  (void)hipMemsetAsync(d_out, 0, (size_t)out_size * sizeof(float), stream);

  const int numTiles = (E + TILE - 1) / TILE;
  if (numTiles <= 0) return;
  // ~2 tiles per block so the double-buffered async pipeline has depth.
  int blocks = numTiles >= 16 ? (numTiles + 1) / 2 : numTiles;
  if (blocks > 16384) blocks = 16384;

  tp_scatter_kernel<<<blocks, THREADS, 0, stream>>>(x, sh, src, dst, out, E,
                                                    numTiles);
}